// net_65309272703155
// MI455X (gfx1250) — compile-verified
//
#include <hip/hip_runtime.h>
#include <math.h>

// Problem constants (fixed by the reference)
#define NN 10000      // nodes
#define BB 16         // batch
#define EE 320000     // edges (excl. self loops)
#define C_IN 16
#define C1 32
#define C2 64
#define C_OUT 8

typedef __attribute__((ext_vector_type(2))) float v2f;
typedef __attribute__((ext_vector_type(8))) float v8f;

// ---------------- setup kernels ----------------

__global__ void zero_i32(int* __restrict__ p, int n) {
    int i = blockIdx.x * blockDim.x + threadIdx.x;
    if (i < n) p[i] = 0;
}

__global__ void count_edges(const int* __restrict__ src, const int* __restrict__ dst,
                            int* __restrict__ outdeg, int* __restrict__ indeg, int E) {
    int e = blockIdx.x * blockDim.x + threadIdx.x;
    if (e < E) {
        atomicAdd(&outdeg[src[e]], 1);
        atomicAdd(&indeg[dst[e]], 1);
    }
}

__global__ void compute_dinv(const int* __restrict__ outdeg, float* __restrict__ dinv, int n) {
    int i = blockIdx.x * blockDim.x + threadIdx.x;
    if (i < n) dinv[i] = rsqrtf((float)(outdeg[i] + 1));   // +1 self loop
}

// single-block exclusive scan of (indeg[i]+1) -> indptr, cursor
__global__ void scan_indptr(const int* __restrict__ indeg, int* __restrict__ indptr,
                            int* __restrict__ cursor, int n) {
    __shared__ int sh[1024];
    __shared__ int carry;
    if (threadIdx.x == 0) carry = 0;
    __syncthreads();
    for (int base = 0; base < n; base += 1024) {
        int i = base + (int)threadIdx.x;
        int v = (i < n) ? (indeg[i] + 1) : 0;   // +1 self loop per node
        sh[threadIdx.x] = v;
        __syncthreads();
        for (int off = 1; off < 1024; off <<= 1) {
            int t = (threadIdx.x >= (unsigned)off) ? sh[threadIdx.x - off] : 0;
            __syncthreads();
            sh[threadIdx.x] += t;
            __syncthreads();
        }
        if (i < n) {
            int excl = carry + sh[threadIdx.x] - v;
            indptr[i] = excl;
            cursor[i] = excl;
        }
        __syncthreads();
        if (threadIdx.x == 0) carry += sh[1023];
        __syncthreads();
    }
    if (threadIdx.x == 0) indptr[n] = carry;
}

__global__ void fill_csr(const int* __restrict__ src, const int* __restrict__ dst,
                         const float* __restrict__ dinv, int* __restrict__ cursor,
                         int* __restrict__ csr_src, float* __restrict__ csr_w,
                         int E, int n) {
    int t = blockIdx.x * blockDim.x + threadIdx.x;
    if (t < E) {
        int s = src[t], d = dst[t];
        int p = atomicAdd(&cursor[d], 1);
        csr_src[p] = s;
        csr_w[p] = dinv[s] * dinv[d];
    } else if (t < E + n) {
        int i = t - E;                      // self loop i -> i
        int p = atomicAdd(&cursor[i], 1);
        csr_src[p] = i;
        csr_w[p] = dinv[i] * dinv[i];
    }
}

// ---------------- dense transform: fp32 WMMA 16x16x4 ----------------
// X: [R, KK] row-major. W: [KK, C] row-major. H: [R, C] (FIRST=true remaps
// (b,n) input rows to (n,b) output rows; since NN%16==0 a tile never crosses
// a batch boundary, so the remapped stores are constant-stride).
// W staged in LDS transposed + zero-padded: inner loop is branch-free,
// 1 global_load_b64 (A) + CT ds_load_b64 (B) + CT WMMAs per K-step.
template <int KK, int C, bool FIRST>
__global__ void node_gemm(const float* __restrict__ X, const float* __restrict__ W,
                          float* __restrict__ H, int R, int Nn, int Bb) {
    constexpr int CT = (C + 15) / 16;
    constexpr int CP = CT * 16;
    constexpr int OSTRIDE = FIRST ? (BB * C) : C;   // float stride between rows v
    __shared__ float sWt[CP * KK];          // [c][k], zero-padded columns
    for (int i = threadIdx.x; i < CP * KK; i += blockDim.x) {
        int c = i / KK;
        int k = i - c * KK;
        sWt[i] = (c < C) ? W[k * C + c] : 0.0f;
    }
    __syncthreads();

    int wave = blockIdx.x * (blockDim.x >> 5) + (threadIdx.x >> 5);
    int lane = threadIdx.x & 31;
    int rowBase = wave * 16;
    if (rowBase >= R) return;               // wave-uniform (grid is exact anyway)

    int m  = lane & 15;                     // A row / B,D column within tile
    int kh = lane >> 4;                     // K-half selector

    v8f acc[CT];
#pragma unroll
    for (int t = 0; t < CT; ++t)
#pragma unroll
        for (int v = 0; v < 8; ++v) acc[t][v] = 0.0f;

    const float* xrow = X + (size_t)(rowBase + m) * KK;

#pragma unroll
    for (int k0 = 0; k0 < KK; k0 += 4) {
        // A fragment (16x4 fp32): lane holds M=m, VGPR0:K=k0+2kh, VGPR1:+1
        v2f a = *reinterpret_cast<const v2f*>(xrow + k0 + 2 * kh);
#pragma unroll
        for (int t = 0; t < CT; ++t) {
            int col = t * 16 + m;
            // B fragment (4x16 fp32) from LDS: contiguous K pair -> ds_load_b64
            v2f b = *reinterpret_cast<const v2f*>(&sWt[col * KK + k0 + 2 * kh]);
            acc[t] = __builtin_amdgcn_wmma_f32_16x16x4_f32(
                false, a, false, b, (short)0, acc[t], false, false);
        }
    }

    // D layout: VGPR v -> row v+8*kh, col = lane&15.
    // One base address per column tile; rows stored at constant stride OSTRIDE.
    size_t rowStart;
    if (FIRST) {
        int b  = rowBase / Nn;                 // wave-uniform (tile within one b)
        int n0 = rowBase - b * Nn + 8 * kh;
        rowStart = (size_t)n0 * Bb + b;        // (n,b) output row of v=0
    } else {
        rowStart = (size_t)(rowBase + 8 * kh);
    }
#pragma unroll
    for (int t = 0; t < CT; ++t) {
        int col = t * 16 + m;
        if (col < C) {                      // compile-time no-op when C%16==0
            float* hp = H + rowStart * C + col;
#pragma unroll
            for (int v = 0; v < 8; ++v)
                hp[v * OSTRIDE] = acc[t][v];   // immediate-offset stores
        }
    }
}

// ---------------- aggregation: CSR gather, one wave per node ----------------
// H: [N, B*C] node-major (c fastest). OUT[i,:] = relu(sum_e w*H[src_e,:] + bias)
template <int NCH4>    // float4 elements per lane; B*C = NCH4*128
__global__ void aggregate(const float* __restrict__ H, const int* __restrict__ indptr,
                          const int* __restrict__ csr_src, const float* __restrict__ csr_w,
                          const float* __restrict__ bias, float* __restrict__ OUT,
                          int Nn, int C) {
    int node = blockIdx.x * (blockDim.x >> 5) + (threadIdx.x >> 5);
    int lane = threadIdx.x & 31;
    if (node >= Nn) return;
    const int BC = NCH4 * 128;

    float4 acc[NCH4];
#pragma unroll
    for (int k = 0; k < NCH4; ++k) acc[k] = make_float4(0.f, 0.f, 0.f, 0.f);

    int e0 = indptr[node], e1 = indptr[node + 1];
    for (int e = e0; e < e1; ++e) {
        int j = csr_src[e];
        float w = csr_w[e];
        const float4* hj = reinterpret_cast<const float4*>(H + (size_t)j * BC);
        // prefetch next edge's row (one 128B line per lane covers the row)
        if (e + 1 < e1) {
            const char* nxt = (const char*)(H + (size_t)csr_src[e + 1] * BC);
            __builtin_prefetch(nxt + lane * 128, 0, 3);
        }
#pragma unroll
        for (int k = 0; k < NCH4; ++k) {
            float4 v = hj[lane + 32 * k];      // coalesced 512B per k
            acc[k].x += w * v.x; acc[k].y += w * v.y;
            acc[k].z += w * v.z; acc[k].w += w * v.w;
        }
    }

    float4* out = reinterpret_cast<float4*>(OUT + (size_t)node * BC);
#pragma unroll
    for (int k = 0; k < NCH4; ++k) {
        int idx4 = lane + 32 * k;
        int c = (idx4 * 4) % C;                // channel of .x (C multiple of 4)
        float4 r;
        r.x = fmaxf(acc[k].x + bias[c + 0], 0.f);
        r.y = fmaxf(acc[k].y + bias[c + 1], 0.f);
        r.z = fmaxf(acc[k].z + bias[c + 2], 0.f);
        r.w = fmaxf(acc[k].w + bias[c + 3], 0.f);
        out[idx4] = r;
    }
}

// ---------------- final linear + sigmoid ----------------
#define NODES_PER_BLOCK 80
__global__ void final_dot(const float* __restrict__ H3, const float* __restrict__ lw,
                          float* __restrict__ acc, int Nn, int Bb) {
    int b = threadIdx.x >> 5;                  // 16 waves, one per batch element
    int lane = threadIdx.x & 31;
    int n0 = blockIdx.x * NODES_PER_BLOCK;
    float s = 0.f;
    for (int i = lane; i < NODES_PER_BLOCK * C_OUT; i += 32) {
        int n = n0 + (i >> 3);                 // C_OUT == 8
        int c = i & 7;
        if (n < Nn) s += H3[((size_t)n * Bb + b) * C_OUT + c] * lw[n * C_OUT + c];
    }
    for (int off = 16; off > 0; off >>= 1) s += __shfl_down(s, off, 32);
    if (lane == 0) atomicAdd(&acc[b], s);
}

__global__ void sigmoid_out(const float* __restrict__ acc, const float* __restrict__ lin_b,
                            float* __restrict__ out, int Bb) {
    int b = threadIdx.x;
    if (b < Bb) out[b] = 1.0f / (1.0f + expf(-(acc[b] + lin_b[0])));
}

// ---------------- launcher ----------------
extern "C" void kernel_launch(void* const* d_in, const int* in_sizes, int n_in,
                              void* d_out, int out_size, void* d_ws, size_t ws_size,
                              hipStream_t stream) {
    const float* x     = (const float*)d_in[0];   // [B,N,C_IN]
    const float* W1    = (const float*)d_in[1];
    const float* b1    = (const float*)d_in[2];
    const float* W2    = (const float*)d_in[3];
    const float* b2    = (const float*)d_in[4];
    const float* W3    = (const float*)d_in[5];
    const float* b3    = (const float*)d_in[6];
    const float* lin_w = (const float*)d_in[7];   // [N*C_OUT]
    const float* lin_b = (const float*)d_in[8];
    const int*   eidx  = (const int*)d_in[9];     // [2,E]
    const int*   src   = eidx;
    const int*   dst   = eidx + EE;
    float* out = (float*)d_out;

    const int R = BB * NN;          // 160000 rows, multiple of 128
    const int E2 = EE + NN;         // edges incl. self loops

    // workspace carve (≈85 MB)
    char* wsp = (char*)d_ws;
    auto carve = [&](size_t bytes) -> void* {
        void* p = (void*)wsp;
        wsp += (bytes + 255) & ~(size_t)255;
        return p;
    };
    float* bufA   = (float*)carve((size_t)NN * BB * 64 * 4);  // max B*C = 1024
    float* bufB   = (float*)carve((size_t)NN * BB * 64 * 4);
    int*   outdeg = (int*)carve((size_t)(2 * NN + 16) * 4);
    int*   indeg  = outdeg + NN;
    float* accB   = (float*)(indeg + NN);
    float* dinv   = (float*)carve((size_t)NN * 4);
    int*   indptr = (int*)carve((size_t)(NN + 1) * 4);
    int*   cursor = (int*)carve((size_t)NN * 4);
    int*   csrs   = (int*)carve((size_t)E2 * 4);
    float* csrw   = (float*)carve((size_t)E2 * 4);

    // ---- graph normalization + CSR build ----
    zero_i32<<<(2 * NN + 16 + 255) / 256, 256, 0, stream>>>(outdeg, 2 * NN + 16);
    count_edges<<<(EE + 255) / 256, 256, 0, stream>>>(src, dst, outdeg, indeg, EE);
    compute_dinv<<<(NN + 255) / 256, 256, 0, stream>>>(outdeg, dinv, NN);
    scan_indptr<<<1, 1024, 0, stream>>>(indeg, indptr, cursor, NN);
    fill_csr<<<(E2 + 255) / 256, 256, 0, stream>>>(src, dst, dinv, cursor, csrs, csrw, EE, NN);

    const int gemmBlocks = R / (16 * 8);     // 1250, exact
    const int aggBlocks  = (NN + 7) / 8;     // 1250, exact

    // ---- layer 1: [B*N,16]@[16,32], store (n,b) ; aggregate B*C=512 ----
    node_gemm<C_IN, C1, true><<<gemmBlocks, 256, 0, stream>>>(x, W1, bufA, R, NN, BB);
    aggregate<4><<<aggBlocks, 256, 0, stream>>>(bufA, indptr, csrs, csrw, b1, bufB, NN, C1);

    // ---- layer 2: [N*B,32]@[32,64] ; aggregate B*C=1024 ----
    node_gemm<C1, C2, false><<<gemmBlocks, 256, 0, stream>>>(bufB, W2, bufA, R, NN, BB);
    aggregate<8><<<aggBlocks, 256, 0, stream>>>(bufA, indptr, csrs, csrw, b2, bufB, NN, C2);

    // ---- layer 3: [N*B,64]@[64,8] ; aggregate B*C=128 ----
    node_gemm<C2, C_OUT, false><<<gemmBlocks, 256, 0, stream>>>(bufB, W3, bufA, R, NN, BB);
    aggregate<1><<<aggBlocks, 256, 0, stream>>>(bufA, indptr, csrs, csrw, b3, bufB, NN, C_OUT);

    // ---- final linear + sigmoid ----
    final_dot<<<(NN + NODES_PER_BLOCK - 1) / NODES_PER_BLOCK, 512, 0, stream>>>(
        bufB, lin_w, accB, NN, BB);
    sigmoid_out<<<1, 16, 0, stream>>>(accB, lin_b, out, BB);
}